// MultiBoxLoss_58093727646073
// MI455X (gfx1250) — compile-verified
//
#include <hip/hip_runtime.h>
#include <hip/hip_bf16.h>

typedef __attribute__((ext_vector_type(2))) float v2f;
typedef __attribute__((ext_vector_type(8))) float v8f;

namespace {
constexpr int B_  = 64;
constexpr int P_  = 8732;
constexpr int C_  = 81;
constexpr int TPR = (P_ + 15) / 16;          // 546 16-prior tiles per batch row
constexpr int NT  = B_ * TPR;                // 34944 wave-tiles total
constexpr int WPB = 8;                       // waves per block (256 thr)
constexpr int BLOCKS_A = (NT + WPB - 1) / WPB;
}

// ---------------------------------------------------------------------------
// K0: zero the per-row positive counters (ws is poisoned, atomics need 0 init)
// ---------------------------------------------------------------------------
__global__ void mb_init_kernel(int* __restrict__ rowNpos) {
    if (threadIdx.x < B_) rowNpos[threadIdx.x] = 0;
}

// ---------------------------------------------------------------------------
// K1: per-prior cross entropy (WMMA row-sum softmax), neg-CE array,
//     per-tile positive-CE / smooth-L1 partials, per-row n_pos.
// One wave (32 lanes) handles 16 priors. Lane l<16 -> prior (l&15), classes
// 4k..4k+1; lane l+16 -> same prior, classes 4k+2..4k+3 (WMMA f32 16x16x4
// A-matrix layout). B = ones => D rows hold sum_k exp(x-max) exactly in f32.
// ---------------------------------------------------------------------------
__global__ __launch_bounds__(256) void mb_ce_kernel(
    const float* __restrict__ cls,  const float* __restrict__ regp,
    const int*   __restrict__ tgt,  const float* __restrict__ regt,
    float* __restrict__ negCE, float* __restrict__ posCEp,
    float* __restrict__ locp,  int* __restrict__ rowNpos)
{
    const int lane = threadIdx.x & 31;
    const int wave = threadIdx.x >> 5;
    const int tile = blockIdx.x * WPB + wave;
    if (tile >= NT) return;                       // wave-uniform exit

    const int b    = tile / TPR;
    const int p0   = (tile % TPR) * 16;
    const int m    = lane & 15;                   // prior row within tile
    const int half = lane >> 4;                   // K-chunk half
    const int p    = p0 + m;
    const bool valid = (p < P_);
    const int  pc    = valid ? p : (P_ - 1);
    const size_t rowbase = ((size_t)b * P_ + pc) * C_;
    const int  t    = valid ? tgt[(size_t)b * P_ + pc] : 0;

    const float NINF = -__builtin_inff();

    // Pass 1: load 42 logits per lane (classes padded to 84), running max,
    // and pick out the target-class logit.
    float xs[42];
    float mx = NINF;
    float xt = 0.f;
#pragma unroll
    for (int k = 0; k < 21; ++k) {
        const int c0 = 4 * k + 2 * half;
        const int c1 = c0 + 1;
        float a0 = cls[rowbase + (c0 < C_ ? c0 : 0)];
        float a1 = cls[rowbase + (c1 < C_ ? c1 : 0)];
        a0 = (valid && c0 < C_) ? a0 : NINF;
        a1 = (valid && c1 < C_) ? a1 : NINF;
        xs[2 * k]     = a0;
        xs[2 * k + 1] = a1;
        mx = fmaxf(mx, fmaxf(a0, a1));
        xt += (valid && c0 == t) ? a0 : 0.f;
        xt += (valid && c1 == t) ? a1 : 0.f;
    }
    mx = fmaxf(mx, __shfl_xor(mx, 16, 32));       // combine lane <-> lane^16
    xt = xt + __shfl_xor(xt, 16, 32);             // exactly one half hit it

    // Pass 2: exp(x - max) fed through WMMA against an all-ones B matrix.
    v8f acc = {0.f, 0.f, 0.f, 0.f, 0.f, 0.f, 0.f, 0.f};
    const v2f bones = {1.f, 1.f};
#pragma unroll
    for (int k = 0; k < 21; ++k) {
        const float e0 = (xs[2 * k]     > NINF) ? __expf(xs[2 * k]     - mx) : 0.f;
        const float e1 = (xs[2 * k + 1] > NINF) ? __expf(xs[2 * k + 1] - mx) : 0.f;
        v2f a = {e0, e1};
        acc = __builtin_amdgcn_wmma_f32_16x16x4_f32(
            false, a, false, bones, (short)0, acc, false, false);
    }

    // Extract this lane's prior row-sum from the D-matrix layout:
    // VGPR r, lanes 0-15 -> rows 0..7; lanes 16-31 -> rows 8..15 (all N equal).
    const int idx = lane & 7;
    float local =
        (idx & 4) ? ((idx & 2) ? ((idx & 1) ? acc[7] : acc[6])
                               : ((idx & 1) ? acc[5] : acc[4]))
                  : ((idx & 2) ? ((idx & 1) ? acc[3] : acc[2])
                               : ((idx & 1) ? acc[1] : acc[0]));
    float other = __shfl_xor(local, 16, 32);
    const bool needOther = (((lane >> 3) & 1) != ((lane >> 4) & 1));
    const float sumexp = needOther ? other : local;

    const float ce  = __logf(sumexp) + mx - xt;   // CE = logsumexp - x_target
    const bool  pos = (t > 0);

    float posce = 0.f, locl = 0.f;
    if (half == 0 && valid) {
        negCE[(size_t)b * P_ + p] = pos ? 0.f : ce;
        if (pos) {
            posce = ce;
            const size_t rb = ((size_t)b * P_ + p) * 4;
#pragma unroll
            for (int j = 0; j < 4; ++j) {
                const float d  = regp[rb + j] - regt[rb + j];
                const float ad = fabsf(d);
                locl += (ad < 1.f) ? 0.5f * d * d : (ad - 0.5f);
            }
        }
    }

    const unsigned long long bm = __ballot(valid && pos);
    const int cnt = __popcll(bm & 0xFFFFull);     // lower-half lanes only

    // Deterministic wave tree reduction (upper lanes contribute 0).
#pragma unroll
    for (int s = 16; s >= 1; s >>= 1) {
        posce += __shfl_xor(posce, s, 32);
        locl  += __shfl_xor(locl,  s, 32);
    }
    if (lane == 0) {
        posCEp[tile] = posce;
        locp[tile]   = locl;
        atomicAdd(&rowNpos[b], cnt);              // integer atomic: deterministic
    }
}

// ---------------------------------------------------------------------------
// K2: per-row sum of top-k negative CE values via 4-pass radix select on the
// (monotone, since CE >= 0) float bit pattern. One block per batch row.
// ---------------------------------------------------------------------------
__global__ __launch_bounds__(256) void mb_topk_kernel(
    const float* __restrict__ negCE, const int* __restrict__ rowNpos,
    float* __restrict__ rowTopk)
{
    const int b = blockIdx.x;
    const float* row = negCE + (size_t)b * P_;
    int k = rowNpos[b] * 3;
    if (k > P_ - 1) k = P_ - 1;

    if (k <= 0) {                                  // uniform across block
        if (threadIdx.x == 0) rowTopk[b] = 0.f;
        return;
    }

    __shared__ unsigned hist[256];
    __shared__ unsigned s_sel, s_rem;
    unsigned prefix = 0;
    unsigned remaining = (unsigned)k;

    for (int pass = 3; pass >= 0; --pass) {
        hist[threadIdx.x] = 0;
        __syncthreads();
        const int sh = pass * 8;
        const unsigned mask_hi = (pass == 3) ? 0u : (0xFFFFFFFFu << (sh + 8));
        for (int i = threadIdx.x; i < P_; i += 256) {
            const unsigned u = __float_as_uint(row[i]);
            if ((u & mask_hi) == (prefix & mask_hi))
                atomicAdd(&hist[(u >> sh) & 255], 1u);
        }
        __syncthreads();
        if (threadIdx.x == 0) {
            unsigned cum = 0;
            int bin = 255;
            for (; bin > 0; --bin) {
                if (cum + hist[bin] >= remaining) break;
                cum += hist[bin];
            }
            s_sel = (unsigned)bin;
            s_rem = remaining - cum;
        }
        __syncthreads();
        prefix |= (s_sel << sh);
        remaining = s_rem;
        __syncthreads();
    }

    // prefix == bit pattern of the k-th largest value T.
    const unsigned T = prefix;
    float    sum = 0.f;
    unsigned cnt = 0;
    for (int i = threadIdx.x; i < P_; i += 256) {
        const float    v = row[i];
        const unsigned u = __float_as_uint(v);
        if (u > T) { sum += v; ++cnt; }
    }
    __shared__ float    ssum[256];
    __shared__ unsigned scnt[256];
    ssum[threadIdx.x] = sum;
    scnt[threadIdx.x] = cnt;
    __syncthreads();
    for (int s = 128; s >= 1; s >>= 1) {
        if ((int)threadIdx.x < s) {
            ssum[threadIdx.x] += ssum[threadIdx.x + s];
            scnt[threadIdx.x] += scnt[threadIdx.x + s];
        }
        __syncthreads();
    }
    if (threadIdx.x == 0) {
        const float tv = __uint_as_float(T);
        rowTopk[b] = ssum[0] + (float)((unsigned)k - scnt[0]) * tv;
    }
}

// ---------------------------------------------------------------------------
// K3: deterministic final reduction + normalization -> (total, cls, loc)
// ---------------------------------------------------------------------------
__global__ __launch_bounds__(256) void mb_final_kernel(
    const float* __restrict__ posCEp, const float* __restrict__ locp,
    const float* __restrict__ rowTopk, const int* __restrict__ rowNpos,
    float* __restrict__ out)
{
    __shared__ float s1[256], s2[256], s3[256];
    __shared__ int   s4[256];
    float a = 0.f, c = 0.f, d = 0.f;
    int np = 0;
    for (int i = threadIdx.x; i < NT; i += 256) { a += posCEp[i]; c += locp[i]; }
    for (int i = threadIdx.x; i < B_; i += 256) { d += rowTopk[i]; np += rowNpos[i]; }
    s1[threadIdx.x] = a; s2[threadIdx.x] = c; s3[threadIdx.x] = d; s4[threadIdx.x] = np;
    __syncthreads();
    for (int s = 128; s >= 1; s >>= 1) {
        if ((int)threadIdx.x < s) {
            s1[threadIdx.x] += s1[threadIdx.x + s];
            s2[threadIdx.x] += s2[threadIdx.x + s];
            s3[threadIdx.x] += s3[threadIdx.x + s];
            s4[threadIdx.x] += s4[threadIdx.x + s];
        }
        __syncthreads();
    }
    if (threadIdx.x == 0) {
        const int   npos  = s4[0];
        const float n     = (float)npos;
        const float safen = fmaxf(n, 1.f);
        const float loc   = (npos == 0) ? 0.f : s2[0] / safen;          // ALPHA=1
        const float clsv  = (npos == 0) ? 0.f : (s1[0] + s3[0]) / safen;
        out[0] = clsv + loc;
        out[1] = clsv;
        out[2] = loc;
    }
}

// ---------------------------------------------------------------------------
extern "C" void kernel_launch(void* const* d_in, const int* in_sizes, int n_in,
                              void* d_out, int out_size, void* d_ws, size_t ws_size,
                              hipStream_t stream) {
    (void)in_sizes; (void)n_in; (void)out_size; (void)ws_size;
    const float* cls  = (const float*)d_in[0];
    const float* regp = (const float*)d_in[1];
    const int*   tgt  = (const int*)d_in[2];
    const float* regt = (const float*)d_in[3];

    float* ws      = (float*)d_ws;
    float* negCE   = ws;                               // B*P
    float* posCEp  = negCE + (size_t)B_ * P_;          // NT
    float* locp    = posCEp + NT;                      // NT
    float* rowTopk = locp + NT;                        // B
    int*   rowNpos = (int*)(rowTopk + B_);             // B
    float* out     = (float*)d_out;

    mb_init_kernel<<<1, 64, 0, stream>>>(rowNpos);
    mb_ce_kernel<<<BLOCKS_A, 256, 0, stream>>>(cls, regp, tgt, regt,
                                               negCE, posCEp, locp, rowNpos);
    mb_topk_kernel<<<B_, 256, 0, stream>>>(negCE, rowNpos, rowTopk);
    mb_final_kernel<<<1, 256, 0, stream>>>(posCEp, locp, rowTopk, rowNpos, out);
}